// LML_15195594293283
// MI455X (gfx1250) — compile-verified
//
#include <hip/hip_runtime.h>

#define NC      4096
#define TOPN    64
#define EPSV    1e-4f
#define NITER   100
#define NBR     10
#define TPB     256
#define EPT     16            // elements per thread = NC / TPB
#define NWAVES  (TPB / 32)    // wave32 on gfx1250

typedef float v4f __attribute__((ext_vector_type(4)));   // native vector for NT stores

#if __has_builtin(__builtin_amdgcn_tanhf)
#define USE_TANH 1
#else
#define USE_TANH 0
#endif

// Full sigmoid (used for the final output values)
__device__ __forceinline__ float full_sigmoid(float t) {
#if USE_TANH
  // sigmoid(t) = 0.5*tanh(t/2) + 0.5   -> v_mul, v_tanh_f32, v_fma
  return __builtin_amdgcn_tanhf(t * 0.5f) * 0.5f + 0.5f;
#else
  // sigmoid(t) = 1/(1+exp2(-t*log2e)) -> v_mul, v_exp_f32, v_add, v_rcp_f32
  float e = __builtin_amdgcn_exp2f(t * -1.4426950408889634f);
  return __builtin_amdgcn_rcpf(1.0f + e);
#endif
}

__global__ __launch_bounds__(TPB) void lml_kernel(const float* __restrict__ x,
                                                  float* __restrict__ out) {
  __shared__ __align__(16) float xs[NC];      // staged row (16 KB)
  __shared__ float    wsum[NWAVES][NBR];      // per-wave branch partial sums
  __shared__ unsigned hist[256];              // radix-select histogram
  __shared__ unsigned wtot[NWAVES];           // per-wave scan totals
  __shared__ unsigned sdig, srr;              // radix pass result broadcast

  const int       tid  = threadIdx.x;
  const int       wid  = tid >> 5;
  const int       lane = tid & 31;
  const long long base = (long long)blockIdx.x * NC;

  // ---- Stage row into LDS with CDNA5 async global->LDS DMA (ASYNCcnt path) ----
  const unsigned lds0 = (unsigned)(size_t)&xs[0];
#pragma unroll
  for (int c = 0; c < 4; ++c) {
    const int idx = c * (TPB * 4) + tid * 4;          // coalesced 16B/lane
    const float* gp = x + base + idx;
    const unsigned lp = lds0 + (unsigned)idx * 4u;
    asm volatile("global_load_async_to_lds_b128 %0, %1, off"
                 :: "v"(lp), "v"(gp) : "memory");
  }
  asm volatile("s_wait_asynccnt 0" ::: "memory");
  __syncthreads();

  // ---- LDS -> registers: 16 floats/thread live in VGPRs for the whole kernel ----
  float    xv[EPT];
  unsigned uv[EPT];
#pragma unroll
  for (int c = 0; c < 4; ++c) {
    const int idx = c * (TPB * 4) + tid * 4;
    v4f v = *(const v4f*)&xs[idx];                    // ds_load_b128
    xv[c * 4 + 0] = v.x; xv[c * 4 + 1] = v.y;
    xv[c * 4 + 2] = v.z; xv[c * 4 + 3] = v.w;
  }
#pragma unroll
  for (int e = 0; e < EPT; ++e) {
    // monotone float->uint map (ascending order preserved)
    unsigned b = __float_as_uint(xv[e]);
    uv[e] = (b & 0x80000000u) ? ~b : (b | 0x80000000u);
  }

  // ---- Exact order statistics: x_sorted[63], x_sorted[64] (descending) ----
  // k-th largest (0-based) == ascending rank (NC-1-k). k=63 -> 4032, k=64 -> 4031.
  float kth[2];
#pragma unroll 1
  for (int q = 0; q < 2; ++q) {
    unsigned rr = (q == 0) ? (unsigned)(NC - TOPN) : (unsigned)(NC - TOPN - 1);
    unsigned prefix = 0;
#pragma unroll
    for (int shift = 24; shift >= 0; shift -= 8) {
      hist[tid] = 0u;                       // TPB == 256 bins
      __syncthreads();
#pragma unroll
      for (int e = 0; e < EPT; ++e) {
        const unsigned u = uv[e];
        const bool ok = (shift == 24) ||
                        ((u >> (shift + 8)) == (prefix >> (shift + 8)));
        if (ok) atomicAdd(&hist[(u >> shift) & 255u], 1u);   // ds_add_u32
      }
      __syncthreads();

      // Block-parallel exclusive prefix over 256 bins (bin = tid):
      const unsigned h = hist[tid];
      unsigned s = h;
#pragma unroll
      for (int off = 1; off < 32; off <<= 1) {   // wave32 inclusive scan
        unsigned t = __shfl_up(s, off, 32);
        if (lane >= off) s += t;
      }
      if (lane == 31) wtot[wid] = s;
      __syncthreads();
      unsigned woff = 0u;
#pragma unroll
      for (int w = 0; w < NWAVES; ++w) woff += (w < wid) ? wtot[w] : 0u;
      const unsigned incl = s + woff;
      const unsigned excl = incl - h;
      if (rr >= excl && rr < incl) {        // exactly one winning bin
        sdig = (unsigned)tid;
        srr  = rr - excl;
      }
      __syncthreads();
      prefix |= sdig << shift;
      rr = srr;
      __syncthreads();                      // hist/wtot reused next pass
    }
    const unsigned bpat =
        (prefix & 0x80000000u) ? (prefix ^ 0x80000000u) : ~prefix;
    kth[q] = __uint_as_float(bpat);
  }

  float lo = -kth[0] - 7.0f;   // -x_sorted[63] - 7
  float hi = -kth[1] + 7.0f;   // -x_sorted[64] + 7

  // ---- 10-way branched bisection on f(nu) = sum_j sigmoid(x_j+nu) - 64 ----
  for (int it = 0; it < NITER; ++it) {
    const float r = hi - lo;
    if (!(r > EPSV)) break;                 // per-row freeze == reference mask

    float cb[NBR];                          // per-branch constant
#pragma unroll
    for (int b = 0; b < NBR; ++b) {
      const float nub = r * ((float)b / 9.0f) + lo;   // linspace(0,1,10) folded
#if USE_TANH
      cb[b] = nub * 0.5f;
#else
      cb[b] = nub;
#endif
    }

    float acc[NBR];
#pragma unroll
    for (int b = 0; b < NBR; ++b) acc[b] = 0.0f;
#pragma unroll
    for (int e = 0; e < EPT; ++e) {
#if USE_TANH
      const float xh = xv[e] * 0.5f;        // 1 mul amortized over 10 branches
#pragma unroll
      for (int b = 0; b < NBR; ++b)         // v_add + v_tanh_f32 + v_add
        acc[b] += __builtin_amdgcn_tanhf(xh + cb[b]);
#else
      const float xe = xv[e];
#pragma unroll
      for (int b = 0; b < NBR; ++b)
        acc[b] += full_sigmoid(xe + cb[b]);
#endif
    }

    // wave32 butterfly reduction (5 steps), then deterministic 8-wave tree
#pragma unroll
    for (int b = 0; b < NBR; ++b) {
      float a = acc[b];
#pragma unroll
      for (int off = 16; off >= 1; off >>= 1)
        a += __shfl_xor(a, off, 32);
      acc[b] = a;
    }
    if (lane == 0) {
#pragma unroll
      for (int b = 0; b < NBR; ++b) wsum[wid][b] = acc[b];
    }
    __syncthreads();

    int neg = 0;
#pragma unroll
    for (int b = 0; b < NBR; ++b) {
      float s = 0.0f;
#pragma unroll
      for (int w = 0; w < NWAVES; ++w) s += wsum[w][b];  // fixed order
#if USE_TANH
      // sum_j sigmoid = 0.5*sum_j tanh + NC/2 ; subtract TOPN
      const float fs = 0.5f * s + (float)(NC / 2 - TOPN);
#else
      const float fs = s - (float)TOPN;
#endif
      neg += (fs < 0.0f) ? 1 : 0;
    }
    __syncthreads();                        // wsum safe for next iteration

    neg -= 1;
    const bool J = (neg < 0);
    const int il = (neg < 0) ? 0 : ((neg > NBR - 2) ? (NBR - 2) : neg);
    const float fl = (float)il;
    float nlo = r * (fl / 9.0f) + lo;       // same rounding path as cb[]
    float nhi = r * ((fl + 1.0f) / 9.0f) + lo;
    if (J) nlo -= 7.0f;
    lo = nlo;
    hi = nhi;
  }

  const float nu = lo + (hi - lo) * 0.5f;

  // ---- Output: sigmoid(x + nu); coalesced non-temporal 16B stores ----
#pragma unroll
  for (int c = 0; c < 4; ++c) {
    const int idx = c * (TPB * 4) + tid * 4;
    v4f o;
    o.x = full_sigmoid(xv[c * 4 + 0] + nu);
    o.y = full_sigmoid(xv[c * 4 + 1] + nu);
    o.z = full_sigmoid(xv[c * 4 + 2] + nu);
    o.w = full_sigmoid(xv[c * 4 + 3] + nu);
    __builtin_nontemporal_store(o, (v4f*)(out + base + idx));
  }
}

extern "C" void kernel_launch(void* const* d_in, const int* in_sizes, int n_in,
                              void* d_out, int out_size, void* d_ws, size_t ws_size,
                              hipStream_t stream) {
  const float* x = (const float*)d_in[0];
  float* out     = (float*)d_out;
  const int nrows = in_sizes[0] / NC;       // 8192 for the reference shapes
  hipLaunchKernelGGL(lml_kernel, dim3(nrows), dim3(TPB), 0, stream, x, out);
}